// YoloV5Loss_63883343561178
// MI455X (gfx1250) — compile-verified
//
#include <hip/hip_runtime.h>
#include <math.h>

// ---------------- constants (match reference) ----------------
#define BATCH   16
#define A_PER   3
#define CNUM    80
#define CH      85            // 5 + C
#define NLAYER  3

// cells per layer = BATCH * A_PER * H * W
#define CELLS0  307200        // 16*3*80*80
#define CELLS1  76800         // 16*3*40*40
#define CELLS2  19200         // 16*3*20*20
#define CELLS_TOT (CELLS0 + CELLS1 + CELLS2)   // 403200, multiple of 256

typedef __attribute__((ext_vector_type(2))) float v2f;
typedef __attribute__((ext_vector_type(8))) float v8f;

__constant__ int   c_dims[NLAYER] = {80, 40, 20};
__constant__ int   c_own_off[NLAYER] = {0, CELLS0, CELLS0 + CELLS1};
// ANCHORS / SCALES, pre-divided
__constant__ float c_anch[NLAYER][A_PER][2] = {
    {{10.f/8.f, 13.f/8.f}, {16.f/8.f, 30.f/8.f}, {33.f/8.f, 23.f/8.f}},
    {{30.f/16.f, 61.f/16.f}, {62.f/16.f, 45.f/16.f}, {59.f/16.f, 119.f/16.f}},
    {{116.f/32.f, 90.f/32.f}, {156.f/32.f, 198.f/32.f}, {373.f/32.f, 326.f/32.f}},
};

__device__ __forceinline__ float softplusf(float x) {
    // stable softplus: max(x,0) + log1p(exp(-|x|))  (== jax.nn.softplus)
    return fmaxf(x, 0.0f) + log1pf(expf(-fabsf(x)));
}
__device__ __forceinline__ float sigmoidf(float x) {
    return 1.0f / (1.0f + expf(-x));
}

// ---------------- kernel 1: re-init workspace every call ----------------
__global__ void yolo_init_ws(int* __restrict__ owners, float* __restrict__ partials,
                             int n_own, int n_part) {
    int i = blockIdx.x * blockDim.x + threadIdx.x;
    if (i < n_own)  owners[i]   = -1;
    if (i < n_part) partials[i] = 0.0f;
}

// ---------------- kernel 2: scatter labels -> owner table + records ----------------
// thread per (layer, label, anchor, candidate); record id = (n*A + a)*5 + c.
// atomicMax on record id is commutative -> deterministic duplicate resolution.
__global__ void yolo_build_targets(const float* __restrict__ labels, int N,
                                   int* __restrict__ owners, float* __restrict__ recs) {
    int t = blockIdx.x * blockDim.x + threadIdx.x;
    int per_layer = N * A_PER * 5;
    if (t >= NLAYER * per_layer) return;
    int layer = t / per_layer;
    int r  = t - layer * per_layer;
    int n  = r / (A_PER * 5);
    int r2 = r - n * (A_PER * 5);
    int a  = r2 / 5;
    int c  = r2 - a * 5;

    const float* L = labels + n * 6;
    int   b   = (int)L[0];
    int   cls = (int)L[1];
    int   W   = c_dims[layer], H = W;
    float lx  = L[2] * (float)W, ly = L[3] * (float)H;
    float lw  = L[4] * (float)W, lh = L[5] * (float)H;
    int   gx  = (int)floorf(lx), gy = (int)floorf(ly);
    float mx  = lx - (float)gx,  my = ly - (float)gy;

    float aw = c_anch[layer][a][0], ah = c_anch[layer][a][1];
    float wr = lw / aw, hr = lh / ah;
    bool rv = (wr > 0.25f) && (wr < 4.0f) && (hr > 0.25f) && (hr < 4.0f);

    bool  cv; float bx, by; int xi = gx, yi = gy;
    switch (c) {
        default:
        case 0: cv = true;                          bx = mx;        by = my;        break;
        case 1: cv = (mx < 0.5f) && (gx > 0);       bx = mx + 1.f;  by = my;        xi = gx - 1; break;
        case 2: cv = (my < 0.5f) && (gy > 0);       bx = mx;        by = my + 1.f;  yi = gy - 1; break;
        case 3: cv = (mx > 0.5f) && (gx < W - 1);   bx = mx - 1.f;  by = my;        xi = gx + 1; break;
        case 4: cv = (my > 0.5f) && (gy < H - 1);   bx = mx;        by = my - 1.f;  yi = gy + 1; break;
    }
    if (!(rv && cv)) return;
    if (b < 0 || b >= BATCH) return;                 // scatter mode='drop'
    if (xi < 0 || xi >= W || yi < 0 || yi >= H) return;

    int rid = (n * A_PER + a) * 5 + c;
    float* rec = recs + (size_t)layer * (size_t)per_layer * 8 + (size_t)rid * 8;
    rec[0] = bx; rec[1] = by; rec[2] = lw; rec[3] = lh; rec[4] = (float)cls;

    int cell = ((b * A_PER + a) * H + yi) * W + xi;
    atomicMax(&owners[c_own_off[layer] + cell], rid);
}

// ---------------- kernel 3: per-cell loss, block partials ----------------
// grid = CELLS_TOT/256 blocks; layer boundaries are multiples of 256 so every
// block is entirely inside one layer. Partial row (16 floats) per block:
// slots [4*layer .. 4*layer+3] = {sum(1-ciou), sum(cls_bce), sum(obj_bce), count}.
__global__ void yolo_cell_loss(const float* __restrict__ p0, const float* __restrict__ p1,
                               const float* __restrict__ p2,
                               const int* __restrict__ owners,
                               const float* __restrict__ recs, int N,
                               float* __restrict__ partials) {
    int g = blockIdx.x * 256 + threadIdx.x;     // g < CELLS_TOT by construction
    int layer, local, W;
    const float* p;
    if (g < CELLS0)               { layer = 0; local = g;                 W = 80; p = p0; }
    else if (g < CELLS0 + CELLS1) { layer = 1; local = g - CELLS0;        W = 40; p = p1; }
    else                          { layer = 2; local = g - CELLS0-CELLS1; W = 20; p = p2; }
    int HW = W * W;
    int x = local % W;
    int y = (local / W) % W;
    int a = (local / HW) % A_PER;
    int b =  local / (HW * A_PER);
    int base = (b * (A_PER * CH) + a * CH) * HW + y * W + x;   // channel stride = HW

    float pobj = p[base + 4 * HW];
    float sp   = softplusf(pobj);

    float vbox = 0.0f, vcls = 0.0f, vobj = sp, vcnt = 0.0f;
    int o = owners[g];
    if (o >= 0) {
        // prefetch the strided class planes (-> global_prefetch_b8) while we do CIoU
        #pragma unroll
        for (int q = 0; q < CNUM; q += 16)
            __builtin_prefetch(&p[base + (5 + q) * HW], 0, 0);

        const float* rec = recs + (size_t)layer * (size_t)(N * A_PER * 5) * 8 + (size_t)o * 8;
        float tx = rec[0], ty = rec[1], tw = rec[2], th = rec[3];
        int   tc = (int)rec[4];

        float rx = p[base + 0 * HW], ry = p[base + 1 * HW];
        float rw = p[base + 2 * HW], rh = p[base + 3 * HW];
        float aw = c_anch[layer][a][0], ah = c_anch[layer][a][1];
        float px = 2.0f * sigmoidf(rx) - 0.5f;
        float py = 2.0f * sigmoidf(ry) - 0.5f;
        float sw = 2.0f * sigmoidf(rw), sh = 2.0f * sigmoidf(rh);
        float pw = sw * sw * aw, ph = sh * sh * ah;

        const float eps = 1e-7f;
        float b1x1 = px - pw * 0.5f, b1x2 = px + pw * 0.5f;
        float b1y1 = py - ph * 0.5f, b1y2 = py + ph * 0.5f;
        float b2x1 = tx - tw * 0.5f, b2x2 = tx + tw * 0.5f;
        float b2y1 = ty - th * 0.5f, b2y2 = ty + th * 0.5f;
        float iw = fmaxf(fminf(b1x2, b2x2) - fmaxf(b1x1, b2x1), 0.0f);
        float ih = fmaxf(fminf(b1y2, b2y2) - fmaxf(b1y1, b2y1), 0.0f);
        float inter = iw * ih;
        float w1 = b1x2 - b1x1, h1 = b1y2 - b1y1 + eps;
        float w2 = b2x2 - b2x1, h2 = b2y2 - b2y1 + eps;
        float uni = w1 * h1 + w2 * h2 - inter + eps;
        float iou = inter / uni;
        float cw  = fmaxf(b1x2, b2x2) - fminf(b1x1, b2x1);
        float chh = fmaxf(b1y2, b2y2) - fminf(b1y1, b2y1);
        float c2  = cw * cw + chh * chh + eps;
        float ddx = (b2x1 + b2x2 - b1x1 - b1x2);
        float ddy = (b2y1 + b2y2 - b1y1 - b1y2);
        float d2  = (ddx * ddx + ddy * ddy) * 0.25f;
        float dat = atanf(w2 / h2) - atanf(w1 / h1);
        const float PIf = 3.14159265358979323846f;
        float v    = (4.0f / (PIf * PIf)) * dat * dat;
        float alpha = v / (v - iou + (1.0f + eps));
        float ciou  = iou - (d2 / c2 + v * alpha);

        vbox = 1.0f - ciou;
        vcnt = 1.0f;
        float tobj = fmaxf(ciou, 0.0f);
        vobj = sp - pobj * tobj;                      // BCE(pobj, tobj)

        float s = 0.0f;
        for (int cc = 0; cc < CNUM; ++cc)
            s += softplusf(p[base + (5 + cc) * HW]);  // BCE(x,0) part
        s -= p[base + (5 + tc) * HW];                 // minus x at one-hot class
        vcls = s;
    }

    // deterministic fixed-order block tree reduction
    __shared__ float sm[4][256];
    sm[0][threadIdx.x] = vbox; sm[1][threadIdx.x] = vcls;
    sm[2][threadIdx.x] = vobj; sm[3][threadIdx.x] = vcnt;
    __syncthreads();
    for (int st = 128; st > 0; st >>= 1) {
        if (threadIdx.x < (unsigned)st) {
            #pragma unroll
            for (int q = 0; q < 4; ++q)
                sm[q][threadIdx.x] += sm[q][threadIdx.x + st];
        }
        __syncthreads();
    }
    if (threadIdx.x == 0) {
        float* row = partials + (size_t)blockIdx.x * 16;
        #pragma unroll
        for (int q = 0; q < 16; ++q) {
            int rel = q - 4 * layer;
            row[q] = (rel >= 0 && rel < 4) ? sm[rel][0] : 0.0f;
        }
    }
}

// ---------------- kernel 4: final reduction (one wave32) ----------------
// Column-sum of the R x 16 partial matrix via V_WMMA_F32_16X16X4_F32 with an
// all-ones A matrix: D[m][n] += sum_k B[k][n]. Because A is all ones, the
// result is invariant to which K-slot each row lands in. EXEC is all ones
// (32 uniform threads) as WMMA requires. f32 in/out -> no precision loss.
__global__ void yolo_final_reduce(const float* __restrict__ partials, int nrows_pad,
                                  float* __restrict__ out) {
    __shared__ float s[16];
    int lane = threadIdx.x;            // 0..31
#if defined(__gfx1250__) && __has_builtin(__builtin_amdgcn_wmma_f32_16x16x4_f32)
    int hi  = lane >> 4;               // 0 for lanes 0-15, 1 for lanes 16-31
    int col = lane & 15;
    v2f ones; ones[0] = 1.0f; ones[1] = 1.0f;
    v8f acc = {};
    for (int base = 0; base < nrows_pad; base += 4) {
        int r0 = base + 2 * hi;        // 4 rows consumed per WMMA (zero-padded tail)
        v2f bm;
        bm[0] = partials[r0 * 16 + col];
        bm[1] = partials[(r0 + 1) * 16 + col];
        acc = __builtin_amdgcn_wmma_f32_16x16x4_f32(
            /*neg_a=*/false, ones, /*neg_b=*/false, bm,
            /*c_mod=*/(short)0, acc, /*reuse_a=*/false, /*reuse_b=*/false);
    }
    if (lane < 16) s[col] = acc[0];    // row M=0 of D = column sums
#else
    if (lane < 16) {                   // deterministic scalar fallback (and host pass)
        float sum = 0.0f;
        for (int r = 0; r < nrows_pad; ++r) sum += partials[r * 16 + lane];
        s[lane] = sum;
    }
#endif
    __syncthreads();
    if (lane == 0) {
        const float objw[3]  = {4.0f, 1.0f, 0.4f};
        const float cells[3] = {(float)CELLS0, (float)CELLS1, (float)CELLS2};
        float boxs = 0.0f, objs = 0.0f, clss = 0.0f;
        for (int l = 0; l < NLAYER; ++l) {
            float bn = s[4 * l + 0], cn = s[4 * l + 1];
            float on = s[4 * l + 2], ct = s[4 * l + 3];
            float denom = fmaxf(ct, 1.0f);
            boxs += bn / denom;
            clss += cn / (denom * (float)CNUM);
            objs += objw[l] * on / cells[l];
        }
        float box = 0.05f * boxs;
        float obj = 1.0f  * objs;
        float cls = 0.5f  * clss;
        float loss = box + obj + cls;
        const float Bf = (float)BATCH;
        out[0] = loss * Bf;            // loss * B
        out[1] = box  * Bf;            // stats[0]
        out[2] = obj  * Bf;            // stats[1]
        out[3] = cls  * Bf;            // stats[2]
        out[4] = loss * Bf;            // stats[3]
    }
}

// ---------------- launch ----------------
extern "C" void kernel_launch(void* const* d_in, const int* in_sizes, int n_in,
                              void* d_out, int out_size, void* d_ws, size_t ws_size,
                              hipStream_t stream) {
    const float* p0     = (const float*)d_in[0];
    const float* p1     = (const float*)d_in[1];
    const float* p2     = (const float*)d_in[2];
    const float* labels = (const float*)d_in[3];
    int N = in_sizes[3] / 6;                       // 300

    const int n_own     = CELLS_TOT;               // owner table (int)
    const int nblocks   = CELLS_TOT / 256;         // 1575 partial rows
    const int nrows_pad = (nblocks + 3) & ~3;      // pad to WMMA K-chunk of 4
    const int n_part    = nrows_pad * 16;

    // ws layout: [owners:int x n_own][records:float x 3*N*15*8][partials:float x n_part]
    int*   owners   = (int*)d_ws;
    float* recs     = (float*)((char*)d_ws + (size_t)n_own * sizeof(int));
    size_t rec_f    = (size_t)NLAYER * (size_t)N * A_PER * 5 * 8;
    float* partials = recs + rec_f;

    int initN = n_own > n_part ? n_own : n_part;
    yolo_init_ws<<<(initN + 255) / 256, 256, 0, stream>>>(owners, partials, n_own, n_part);

    int bt = NLAYER * N * A_PER * 5;               // 13,500 candidate entries
    yolo_build_targets<<<(bt + 255) / 256, 256, 0, stream>>>(labels, N, owners, recs);

    yolo_cell_loss<<<nblocks, 256, 0, stream>>>(p0, p1, p2, owners, recs, N, partials);

    yolo_final_reduce<<<1, 32, 0, stream>>>(partials, nrows_pad, (float*)d_out);
}